// FH_9045201126061
// MI455X (gfx1250) — compile-verified
//
#include <hip/hip_runtime.h>

// FitzHugh-Nagumo Euler scan: z (B,T,L) f32 -> V/2 (B,T,L) f32.
// Sequential over T=2000, parallel over B*L = 16384 lanes (512 wave32).
// Memory-bound: 262 MB streamed => ~11us floor at 23.3 TB/s.
//
// Per-wave budget: one 128B line read + one written per 5.7ns; serial FMA
// chain per step ~12 cycles (~4.8ns) -> memory latency must be fully hidden:
//  - triple-buffered register pipeline: 32 NT loads in flight per wave
//    (~180ns tolerance, covers L2-hit latency).
//  - global_prefetch_b8 at 256-step distance: ~16.8 MB HBM->L2 in flight
//    chip-wide (above the ~14 MB BW x latency product).
//  - non-temporal loads/stores: 262 MB read-once/write-once streams must
//    not thrash the 192 MB L2.
//  - scalar-uniform prefetch guard: one s_cmp/s_cbranch per 16-step block.

#define B_ 16
#define T_ 2000
#define L_ 1024

namespace {
constexpr float DT = 0.1f;
constexpr int U  = 16;   // block size; 3 buffers -> 32 loads in flight
constexpr int PF = 256;  // prefetch distance in timesteps (multiple of U)

__device__ __forceinline__ void fh_step(float& v, float& s, float zk, float* outp) {
  const float vn = (1.0f + DT) * v - (DT / 3.0f) * (v * v * v) - DT * s + DT * zk;
  const float sn = s + (DT * 0.08f) * (v + 0.7f - 0.8f * s);
  v = vn;
  s = sn;
  __builtin_nontemporal_store(0.5f * v, outp);
}

__device__ __forceinline__ void load_block(float (&buf)[U], const float* zp, int k0) {
#pragma unroll
  for (int i = 0; i < U; ++i)
    buf[i] = __builtin_nontemporal_load(zp + (k0 + i) * L_);
}

__device__ __forceinline__ void pf_block(const float* zp, int k) {
  if (k + PF + U - 1 < T_) {  // scalar-uniform: one branch covers 16 prefetches
#pragma unroll
    for (int i = 0; i < U; ++i)
      __builtin_prefetch(zp + (k + PF + i) * L_, 0, 0);
  }
}

__device__ __forceinline__ void consume_block(const float (&buf)[U], float& v, float& s,
                                              float* op, int k) {
#pragma unroll
  for (int i = 0; i < U; ++i)
    fh_step(v, s, buf[i], op + (k + 1 + i) * L_);
}

__global__ __launch_bounds__(128, 1) void fh_scan_kernel(const float* __restrict__ z,
                                                         float* __restrict__ out) {
  const int g = blockIdx.x * 128 + threadIdx.x;   // 0..16383 (grid exact)
  const int b = g >> 10;                          // block-uniform (8 blocks per b)
  const int l = g & (L_ - 1);
  const float* zp = z   + (size_t)b * T_ * L_ + l;
  float*       op = out + (size_t)b * T_ * L_ + l;

  float v = 0.0f, s = 0.0f;

  // V[:,0] = 0
  __builtin_nontemporal_store(0.0f, op);

  float A[U], B[U], C[U];

  // Prologue: A = z[0..15], B = z[16..31]  (32 loads in flight)
  load_block(A, zp, 0);
  load_block(B, zp, U);

  int k = 0;
  // Invariant at loop head: A = z[k..k+15], B = z[k+16..k+31].
  // 40 iterations x 48 steps = 1920 steps; loads reach z[1951].
  for (int m = 0; m < 40; ++m) {
    pf_block(zp, k);
    load_block(C, zp, k + 2 * U);        // z[k+32..k+47]
    consume_block(A, v, s, op, k);       // steps k..k+15
    k += U;

    pf_block(zp, k);
    load_block(A, zp, k + 2 * U);
    consume_block(B, v, s, op, k);
    k += U;

    pf_block(zp, k);
    load_block(B, zp, k + 2 * U);
    consume_block(C, v, s, op, k);
    k += U;
  }

  // Tail: k = 1920, A = z[1920..1935], B = z[1936..1951]; 79 steps remain.
  load_block(C, zp, k + 2 * U);          // z[1952..1967]
  consume_block(A, v, s, op, k);         // 1920..1935
  k += U;

  load_block(A, zp, k + 2 * U);          // z[1968..1983]
  consume_block(B, v, s, op, k);         // 1936..1951
  k += U;

  load_block(B, zp, k + 2 * U);          // z[1984..1999] (last in-bounds line)
  consume_block(C, v, s, op, k);         // 1952..1967
  k += U;

  consume_block(A, v, s, op, k);         // 1968..1983
  k += U;

  // Final 15 steps: z[1984..1998] -> out[1985..1999]
#pragma unroll
  for (int i = 0; i < U - 1; ++i)
    fh_step(v, s, B[i], op + (k + 1 + i) * L_);
}
}  // namespace

extern "C" void kernel_launch(void* const* d_in, const int* in_sizes, int n_in,
                              void* d_out, int out_size, void* d_ws, size_t ws_size,
                              hipStream_t stream) {
  (void)in_sizes; (void)n_in; (void)out_size; (void)d_ws; (void)ws_size;
  const float* z = (const float*)d_in[0];
  float* out = (float*)d_out;
  const int total = B_ * L_;            // 16384 lanes
  dim3 grid(total / 128), block(128);   // 128 blocks x 4 wave32
  hipLaunchKernelGGL(fh_scan_kernel, grid, block, 0, stream, z, out);
}